// SkipGCN_52656299049172
// MI455X (gfx1250) — compile-verified
//
#include <hip/hip_runtime.h>

typedef __attribute__((ext_vector_type(2))) float v2f;
typedef __attribute__((ext_vector_type(8))) float v8f;

// ---------------------------------------------------------------- utilities
__global__ void k_zero_f(float* __restrict__ p, long long n) {
    long long i = (long long)blockIdx.x * blockDim.x + threadIdx.x;
    long long stride = (long long)gridDim.x * blockDim.x;
    for (; i < n; i += stride) p[i] = 0.0f;
}

__global__ void k_fill_i(int* __restrict__ p, int n, int v) {
    int i = blockIdx.x * blockDim.x + threadIdx.x;
    if (i < n) p[i] = v;
}

// ------------------------------------------------------- degree / normalizer
__global__ void k_deg(const int* __restrict__ dst, int E, int* __restrict__ deg) {
    int e = blockIdx.x * blockDim.x + threadIdx.x;
    if (e < E) atomicAdd(&deg[dst[e]], 1);
}

__global__ void k_dinv(const int* __restrict__ deg, float* __restrict__ dinv, int n) {
    int i = blockIdx.x * blockDim.x + threadIdx.x;
    if (i < n) dinv[i] = rsqrtf((float)deg[i]);   // deg >= 1 (self loop)
}

// --------------------------------------------------- h1 = x @ W1 via f32 WMMA
// x: [N,165] row-major, W1: [165,128] row-major, h1: [N,128]
// 32 rows per block, 8 waves; wave w owns 16-col tile w and both 16-row tiles.
// K loop is peeled: 41 unguarded 4-wide steps + one partial tail tile.
#define GEMM1_ROWS 32
__global__ __launch_bounds__(256) void k_gemm1(const float* __restrict__ x,
                                               const float* __restrict__ W1,
                                               float* __restrict__ h1, int Nn) {
    __shared__ float Xl[GEMM1_ROWS * 168];   // K padded 165 -> 168 with zeros
    const int tid = threadIdx.x;
    const long long rowBase = (long long)blockIdx.x * GEMM1_ROWS;

    for (int i = tid; i < GEMM1_ROWS * 168; i += 256) {
        int r = i / 168;
        int kk = i - r * 168;
        long long row = rowBase + r;
        Xl[i] = (kk < 165 && row < Nn) ? x[row * 165 + kk] : 0.0f;
    }
    __syncthreads();

    const int wave = tid >> 5;
    const int lane = tid & 31;
    const int hi   = lane >> 4;      // which K half-pair this lane holds
    const int lo   = lane & 15;      // M (for A) / N (for B) within tile
    const int colBase = wave << 4;   // 8 waves * 16 = 128 cols

    const float* x0 = &Xl[lo * 168];
    const float* x1 = &Xl[(16 + lo) * 168];
    const float* Wc = W1 + colBase + lo;

    v8f c0 = {};
    v8f c1 = {};
    // main loop: k = 0..160 step 4 -> max row read = 163 < 165, no guards
    for (int k = 0; k < 164; k += 4) {
        const int ka = k + (hi << 1);
        v2f b;  b.x  = Wc[ka * 128];  b.y  = Wc[(ka + 1) * 128];
        v2f a0; a0.x = x0[ka];        a0.y = x0[ka + 1];
        v2f a1; a1.x = x1[ka];        a1.y = x1[ka + 1];
        c0 = __builtin_amdgcn_wmma_f32_16x16x4_f32(false, a0, false, b,
                                                   (short)0, c0, false, false);
        c1 = __builtin_amdgcn_wmma_f32_16x16x4_f32(false, a1, false, b,
                                                   (short)0, c1, false, false);
    }
    // peeled tail: K-tile covering rows 164..167 (only row 164 is real)
    {
        const int ka = 164 + (hi << 1);
        v2f b;  b.x  = (hi == 0) ? Wc[164 * 128] : 0.0f;  b.y = 0.0f;
        v2f a0; a0.x = x0[ka];        a0.y = x0[ka + 1];  // LDS zero-padded
        v2f a1; a1.x = x1[ka];        a1.y = x1[ka + 1];
        c0 = __builtin_amdgcn_wmma_f32_16x16x4_f32(false, a0, false, b,
                                                   (short)0, c0, false, false);
        c1 = __builtin_amdgcn_wmma_f32_16x16x4_f32(false, a1, false, b,
                                                   (short)0, c1, false, false);
    }

    // C/D layout: VGPR r -> M = r + 8*hi, N = lane&15
    long long r0 = rowBase + hi * 8;
    long long r1 = rowBase + 16 + hi * 8;
#pragma unroll
    for (int r = 0; r < 8; ++r) {
        if (r0 + r < Nn) h1[(r0 + r) * 128 + colBase + lo] = c0[r];
        if (r1 + r < Nn) h1[(r1 + r) * 128 + colBase + lo] = c1[r];
    }
}

// ------------------------------- edge scatter: agg[dst] += norm * h1[src]
// one thread per (edge, 4 features): b128 gathers, wave-uniform edge indices
__global__ void k_agg1(const int* __restrict__ src, const int* __restrict__ dst,
                       const float* __restrict__ dinv, const float* __restrict__ h1,
                       float* __restrict__ agg, long long total /* E*32 */) {
    long long idx = (long long)blockIdx.x * 256 + threadIdx.x;
    if (idx >= total) return;
    long long e = idx >> 5;
    int f4 = (int)(idx & 31) << 2;
    int s = src[e], d = dst[e];
    float nrm = dinv[s] * dinv[d];
    const float4 v = *(const float4*)(h1 + (long long)s * 128 + f4);
    float* o = agg + (long long)d * 128 + f4;
    atomicAdd(o + 0, v.x * nrm);
    atomicAdd(o + 1, v.y * nrm);
    atomicAdd(o + 2, v.z * nrm);
    atomicAdd(o + 3, v.w * nrm);
}

// self-loop + bias + relu, in place into h1 (float4, one thread per node/4feat)
__global__ void k_relu(float* __restrict__ h1, const float* __restrict__ agg,
                       const float* __restrict__ dinv, const float* __restrict__ b1,
                       long long total /* N*32 */) {
    long long idx = (long long)blockIdx.x * 256 + threadIdx.x;
    if (idx >= total) return;
    long long i = idx >> 5;
    int f4 = (int)(idx & 31) << 2;
    float dv = dinv[i];
    float d2 = dv * dv;
    long long base = i * 128 + f4;
    float4 a = *(const float4*)(agg + base);
    float4 h = *(const float4*)(h1 + base);
    float4 bb = *(const float4*)(b1 + f4);
    float4 r;
    r.x = a.x + h.x * d2 + bb.x;
    r.y = a.y + h.y * d2 + bb.y;
    r.z = a.z + h.z * d2 + bb.z;
    r.w = a.w + h.w * d2 + bb.w;
    r.x = r.x > 0.0f ? r.x : 0.0f;
    r.y = r.y > 0.0f ? r.y : 0.0f;
    r.z = r.z > 0.0f ? r.z : 0.0f;
    r.w = r.w > 0.0f ? r.w : 0.0f;
    *(float4*)(h1 + base) = r;
}

// ----------------------------------------- h2 = h1 @ W2  (W2: [128,2], skinny)
__global__ __launch_bounds__(256) void k_gemm2(const float* __restrict__ hr,
                                               const float* __restrict__ W2,
                                               float* __restrict__ h2, int Nn) {
    __shared__ float Wl[256];
    int tid = threadIdx.x;
    Wl[tid] = W2[tid];            // 128*2 floats == 256 threads
    __syncthreads();
    int wave = tid >> 5, lane = tid & 31;
    long long i = (long long)blockIdx.x * 8 + wave;
    if (i >= Nn) return;
    const float* row = hr + i * 128;
    float s0 = 0.0f, s1 = 0.0f;
    for (int k = lane; k < 128; k += 32) {
        float v = row[k];
        s0 += v * Wl[2 * k];
        s1 += v * Wl[2 * k + 1];
    }
    for (int off = 16; off > 0; off >>= 1) {
        s0 += __shfl_xor(s0, off, 32);
        s1 += __shfl_xor(s1, off, 32);
    }
    if (lane == 0) { h2[i * 2] = s0; h2[i * 2 + 1] = s1; }
}

// --------------------- second edge scatter into d_out: one thread per edge
__global__ void k_agg2(const int* __restrict__ src, const int* __restrict__ dst,
                       const float* __restrict__ dinv, const float* __restrict__ h2,
                       float* __restrict__ out, int E) {
    long long e = (long long)blockIdx.x * 256 + threadIdx.x;
    if (e >= E) return;
    int s = src[e], d = dst[e];
    float nrm = dinv[s] * dinv[d];
    float2 v = *(const float2*)(h2 + (long long)s * 2);
    atomicAdd(&out[(long long)d * 2 + 0], v.x * nrm);
    atomicAdd(&out[(long long)d * 2 + 1], v.y * nrm);
}

// ---------------- out += self-loop*h2 + b2 + x@Ws + bs   (Ws: [165,2], skinny)
__global__ __launch_bounds__(256) void k_final(const float* __restrict__ x,
                                               const float* __restrict__ Ws,
                                               const float* __restrict__ h2,
                                               const float* __restrict__ dinv,
                                               const float* __restrict__ b2,
                                               const float* __restrict__ bs,
                                               float* __restrict__ out, int Nn) {
    __shared__ float Wl[336];
    int tid = threadIdx.x;
    for (int i = tid; i < 330; i += 256) Wl[i] = Ws[i];
    __syncthreads();
    int wave = tid >> 5, lane = tid & 31;
    long long i = (long long)blockIdx.x * 8 + wave;
    if (i >= Nn) return;
    const float* row = x + i * 165;
    float s0 = 0.0f, s1 = 0.0f;
    for (int k = lane; k < 165; k += 32) {
        float v = row[k];
        s0 += v * Wl[2 * k];
        s1 += v * Wl[2 * k + 1];
    }
    for (int off = 16; off > 0; off >>= 1) {
        s0 += __shfl_xor(s0, off, 32);
        s1 += __shfl_xor(s1, off, 32);
    }
    if (lane == 0) {
        float dv = dinv[i];
        float d2 = dv * dv;
        out[i * 2 + 0] += h2[i * 2 + 0] * d2 + b2[0] + s0 + bs[0];
        out[i * 2 + 1] += h2[i * 2 + 1] * d2 + b2[1] + s1 + bs[1];
    }
}

// ------------------------------------------------------------------- launcher
extern "C" void kernel_launch(void* const* d_in, const int* in_sizes, int n_in,
                              void* d_out, int out_size, void* d_ws, size_t ws_size,
                              hipStream_t stream) {
    (void)n_in; (void)out_size; (void)ws_size;
    const float* x  = (const float*)d_in[0];
    const int*   ei = (const int*)d_in[1];     // int32 (JAX x64 disabled)
    const float* W1 = (const float*)d_in[2];
    const float* b1 = (const float*)d_in[3];
    const float* W2 = (const float*)d_in[4];
    const float* b2 = (const float*)d_in[5];
    const float* Ws = (const float*)d_in[6];
    const float* bs = (const float*)d_in[7];
    float* out = (float*)d_out;

    const int Nn = in_sizes[0] / 165;
    const int E  = in_sizes[1] / 2;
    const int* src = ei;
    const int* dst = ei + E;

    // workspace carving (256B aligned)
    char* ws = (char*)d_ws;
    size_t off = 0;
    auto take = [&](size_t bytes) -> void* {
        void* p = (void*)(ws + off);
        off += (bytes + 255) & ~(size_t)255;
        return p;
    };
    int*   deg  = (int*)  take((size_t)Nn * 4);
    float* dinv = (float*)take((size_t)Nn * 4);
    float* h1   = (float*)take((size_t)Nn * 128 * 4);
    float* agg1 = (float*)take((size_t)Nn * 128 * 4);
    float* h2   = (float*)take((size_t)Nn * 2 * 4);

    const long long tot1v = (long long)E * 32;   // edge scatter, float4 lanes
    const long long totRv = (long long)Nn * 32;  // relu pass, float4 lanes
    const long long totR  = (long long)Nn * 128;

    // init
    k_fill_i<<<(Nn + 255) / 256, 256, 0, stream>>>(deg, Nn, 1);
    k_zero_f<<<50000, 256, 0, stream>>>(agg1, totR);
    k_zero_f<<<(Nn * 2 + 255) / 256, 256, 0, stream>>>(out, (long long)Nn * 2);

    // normalization
    k_deg<<<(E + 255) / 256, 256, 0, stream>>>(dst, E, deg);
    k_dinv<<<(Nn + 255) / 256, 256, 0, stream>>>(deg, dinv, Nn);

    // layer 1
    k_gemm1<<<(Nn + GEMM1_ROWS - 1) / GEMM1_ROWS, 256, 0, stream>>>(x, W1, h1, Nn);
    k_agg1<<<(unsigned)((tot1v + 255) / 256), 256, 0, stream>>>(src, dst, dinv, h1, agg1, tot1v);
    k_relu<<<(unsigned)((totRv + 255) / 256), 256, 0, stream>>>(h1, agg1, dinv, b1, totRv);

    // layer 2 + skip
    k_gemm2<<<(Nn + 7) / 8, 256, 0, stream>>>(h1, W2, h2, Nn);
    k_agg2<<<(E + 255) / 256, 256, 0, stream>>>(src, dst, dinv, h2, out, E);
    k_final<<<(Nn + 7) / 8, 256, 0, stream>>>(x, Ws, h2, dinv, b2, bs, out, Nn);
}